// AdaptiveFanOut_57037165691068
// MI455X (gfx1250) — compile-verified
//
#include <hip/hip_runtime.h>
#include <hip/hip_bf16.h>

#define BB 8
#define NEW_S 2048
#define RES_S 4096
#define HDIM 1024

typedef __attribute__((ext_vector_type(4))) float v4f;

// ---------------------------------------------------------------------------
// K1: fill the inverse map with -1 (re-done every call; ws is not re-poisoned)
// ---------------------------------------------------------------------------
__global__ __launch_bounds__(256) void afo_fill_inv(int* __restrict__ inv, int n) {
    int i = blockIdx.x * blockDim.x + threadIdx.x;
    if (i < n) inv[i] = -1;
}

// ---------------------------------------------------------------------------
// K2: per-batch prefix-sum of counts -> inverse scatter map.
// One 256-thread block per batch; each thread scans 8 contiguous counts,
// block-wide Hillis-Steele scan over thread totals in LDS, min-reduce of the
// first non-positive count ("break" semantics of the reference loop).
// Valid tokens have strictly increasing cumsum -> unique targets -> plain store.
// ---------------------------------------------------------------------------
__global__ __launch_bounds__(256) void afo_scan(const int* __restrict__ counts,
                                                int* __restrict__ inv) {
    const int b = blockIdx.x;
    const int t = threadIdx.x;
    __shared__ int sums[256];
    __shared__ int firstZero;

    const int* c = counts + b * NEW_S;
    int local[8];
    int s = 0;
    int myFz = NEW_S;
#pragma unroll
    for (int k = 0; k < 8; ++k) {
        int i = t * 8 + k;
        int v = c[i];
        if (v <= 0 && myFz == NEW_S) myFz = i;
        s += v;
        local[k] = s;  // inclusive within this thread's chunk
    }
    sums[t] = s;
    if (t == 0) firstZero = NEW_S;
    __syncthreads();
    if (myFz < NEW_S) atomicMin(&firstZero, myFz);

    // inclusive scan over per-thread sums
    for (int off = 1; off < 256; off <<= 1) {
        int v = (t >= off) ? sums[t - off] : 0;
        __syncthreads();
        if (t >= off) sums[t] += v;
        __syncthreads();
    }
    const int excl = (t == 0) ? 0 : sums[t - 1];
    const int fz = firstZero;

#pragma unroll
    for (int k = 0; k < 8; ++k) {
        int i = t * 8 + k;
        int idx = excl + local[k] - 1;           // cumsum - 1
        if (i < fz && idx >= 0 && idx < RES_S)   // mode='drop' OOB guard
            inv[b * RES_S + idx] = i;
    }
}

// ---------------------------------------------------------------------------
// K3: gather. One block per output row (b, r): 4 KB residual row is staged
// into LDS via gfx1250 async global->LDS B128 (ASYNCcnt), overlapped with the
// inv lookup + dependent hidden-row load (non-temporal). Output store is NT.
// ---------------------------------------------------------------------------
__global__ __launch_bounds__(256) void afo_gather(const float* __restrict__ hidden,
                                                  const float* __restrict__ residual,
                                                  const int* __restrict__ inv,
                                                  float* __restrict__ out) {
    __shared__ v4f buf[256];               // one 4 KB row
    const int row = blockIdx.x;            // row = b*RES_S + r
    const int t = threadIdx.x;
    const int b = row >> 12;               // RES_S == 4096
    const size_t rowBase = (size_t)row * HDIM;

    // Kick off the async residual-row fetch: each lane moves 16 B to LDS.
    {
        const float* gsrc = residual + rowBase + t * 4;
        unsigned ldsOff = (unsigned)(size_t)(&buf[t]);  // low 32 bits = LDS offset
        asm volatile("global_load_async_to_lds_b128 %0, %1, off"
                     :: "v"(ldsOff), "v"(gsrc)
                     : "memory");
    }

    // Meanwhile resolve the scatter source for this row and fetch it (if any).
    const int src = inv[row];
    v4f add = (v4f)(0.0f);
    if (src >= 0) {
        const v4f* hrow = (const v4f*)(hidden + ((size_t)b * NEW_S + (size_t)src) * HDIM);
        add = __builtin_nontemporal_load(hrow + t);
    }

    // Wait for this wave's async transfers; each lane reads back its own 16 B.
    asm volatile("s_wait_asynccnt 0" ::: "memory");
    v4f v = buf[t];
    v += add;

    v4f* orow = (v4f*)(out + rowBase);
    __builtin_nontemporal_store(v, orow + t);
}

// ---------------------------------------------------------------------------
// kernel_launch
//   d_in[0] hidden_states            float32 [8, 2048, 1024]
//   d_in[1] attention_mask           int     (unused by reference)
//   d_in[2] merged_embeddings_counts int     [8, 2048]
//   d_in[3] residual_hidden_states   float32 [8, 4096, 1024]
//   d_in[4] residual_attention_mask  int     (unused by reference)
//   d_out   restored                 float32 [8, 4096, 1024]
//   d_ws    inv map                  int     [8, 4096]  (128 KB)
// ---------------------------------------------------------------------------
extern "C" void kernel_launch(void* const* d_in, const int* in_sizes, int n_in,
                              void* d_out, int out_size, void* d_ws, size_t ws_size,
                              hipStream_t stream) {
    const float* hidden   = (const float*)d_in[0];
    const int*   counts   = (const int*)d_in[2];
    const float* residual = (const float*)d_in[3];
    float*       out      = (float*)d_out;
    int*         inv      = (int*)d_ws;

    const int nInv = BB * RES_S;
    afo_fill_inv<<<(nInv + 255) / 256, 256, 0, stream>>>(inv, nInv);
    afo_scan<<<BB, 256, 0, stream>>>(counts, inv);
    afo_gather<<<BB * RES_S, 256, 0, stream>>>(hidden, residual, inv, out);
}